// SkeLossMultiClass_90125593739964
// MI455X (gfx1250) — compile-verified
//
#include <hip/hip_runtime.h>

// Problem constants (B,C,D,H,W) = (2,4,64,256,256)
#define C_    4
#define DHW   4194304   // 64*256*256
#define NV4   (DHW/4)   // 1048576 float4 / int4 elements per (b, plane)
#define BETA  25.0f
#define EPS   1e-4f

typedef float v2f __attribute__((ext_vector_type(2)));
typedef float v8f __attribute__((ext_vector_type(8)));

// Full-precision 32-lane sum using V_WMMA_F32_16X16X4_F32.
// A = per-lane {v, 0}  (A[m][0]=v[m], A[m][1]=0, A[m][2]=v[m+16], A[m][3]=0)
// B = ones             => D[m][n] = v[m] + v[m+16] for every n.
// Per-lane sum of the 8 D VGPRs gives rows 0..7 (lanes 0-15) / rows 8..15
// (lanes 16-31); one xor-16 shuffle completes the 32-lane sum (all lanes).
// Requires EXEC == all ones (guaranteed: called only at uniform points).
__device__ __forceinline__ float wave_sum32(float v) {
  v2f a;    a[0] = v;    a[1] = 0.0f;
  v2f ones; ones[0] = 1.0f; ones[1] = 1.0f;
  v8f c = {};
  c = __builtin_amdgcn_wmma_f32_16x16x4_f32(false, a, false, ones,
                                            (short)0, c, false, false);
  float r = c[0] + c[1] + c[2] + c[3] + c[4] + c[5] + c[6] + c[7];
  r += __shfl_xor(r, 16, 32);
  return r;
}

// Per-voxel contribution for the 3 foreground classes.
// acc layout: [class*3 + {num,den1,den2}]
__device__ __forceinline__ void accum3(int ves, int ske,
                                       float p1, float p2, float p3,
                                       float acc[9]) {
  const float pr[3] = {p1, p2, p3};
#pragma unroll
  for (int ci = 0; ci < 3; ++ci) {
    const int c  = ci + 1;
    const float sw  = (ske == c) ? 1.1f : 0.1f;   // 0.1 + ske_i
    const bool  vm  = (ves == c);
    const float t   = vm ? sw : 0.0f;             // ves_i * importance
    const float imp = vm ? sw : (sw + 1.0f);      // 1.1 - ves_i + ske_i
    acc[ci * 3 + 0] = fmaf(pr[ci], t,   acc[ci * 3 + 0]);  // num
    acc[ci * 3 + 1] = fmaf(pr[ci], imp, acc[ci * 3 + 1]);  // den1
    acc[ci * 3 + 2] += t;                                  // den2
  }
}

extern "C" __global__ void __launch_bounds__(256)
ske_partial_kernel(const float* __restrict__ output,
                   const int*   __restrict__ labels,
                   float* __restrict__ partial, int P) {
  const int b = blockIdx.y;
  const float4* __restrict__ p1 =
      reinterpret_cast<const float4*>(output + (size_t)b * C_ * DHW + (size_t)1 * DHW);
  const float4* __restrict__ p2 =
      reinterpret_cast<const float4*>(output + (size_t)b * C_ * DHW + (size_t)2 * DHW);
  const float4* __restrict__ p3 =
      reinterpret_cast<const float4*>(output + (size_t)b * C_ * DHW + (size_t)3 * DHW);
  const int4* __restrict__ vv =
      reinterpret_cast<const int4*>(labels + (size_t)b * 2 * DHW);
  const int4* __restrict__ kk =
      reinterpret_cast<const int4*>(labels + (size_t)b * 2 * DHW + (size_t)DHW);

  float acc[9];
#pragma unroll
  for (int i = 0; i < 9; ++i) acc[i] = 0.0f;

  // stride is a multiple of 256 and NV4 % 256 == 0  =>  every thread in a
  // block runs the same trip count: EXEC stays all-ones (WMMA-safe).
  const int stride = P * 256;
  for (int i = blockIdx.x * 256 + threadIdx.x; i < NV4; i += stride) {
    const int4   v  = vv[i];
    const int4   k  = kk[i];
    const float4 a  = p1[i];
    const float4 bb = p2[i];
    const float4 cc = p3[i];
    accum3(v.x, k.x, a.x, bb.x, cc.x, acc);
    accum3(v.y, k.y, a.y, bb.y, cc.y, acc);
    accum3(v.z, k.z, a.z, bb.z, cc.z, acc);
    accum3(v.w, k.w, a.w, bb.w, cc.w, acc);
  }

  // Wave-level reduction via WMMA (uniform control flow here), then LDS
  // across the 8 waves of the block, then one partial record per block.
  __shared__ float smem[8][9];
  const int wave = threadIdx.x >> 5;
  const int lane = threadIdx.x & 31;

  float wsum[9];
#pragma unroll
  for (int q = 0; q < 9; ++q) wsum[q] = wave_sum32(acc[q]);

  if (lane == 0) {
#pragma unroll
    for (int q = 0; q < 9; ++q) smem[wave][q] = wsum[q];
  }
  __syncthreads();

  if (threadIdx.x < 9) {
    float s = 0.0f;
#pragma unroll
    for (int w = 0; w < 8; ++w) s += smem[w][threadIdx.x];
    partial[((size_t)b * P + blockIdx.x) * 9 + threadIdx.x] = s;
  }
}

extern "C" __global__ void __launch_bounds__(576)
ske_final_kernel(const float* __restrict__ partial, int P,
                 float* __restrict__ out) {
  __shared__ float s[18];
  const int g    = threadIdx.x >> 5;  // 18 waves: (b, quantity) pairs
  const int lane = threadIdx.x & 31;
  const int b    = g / 9;
  const int r    = g % 9;

  float acc = 0.0f;
  for (int i = lane; i < P; i += 32)
    acc += partial[((size_t)b * P + i) * 9 + r];

  const float tot = wave_sum32(acc);   // reconverged: full EXEC
  if (lane == 0) s[g] = tot;
  __syncthreads();

  if (threadIdx.x == 0) {
    float loss = 0.0f;
#pragma unroll
    for (int bb = 0; bb < 2; ++bb) {
#pragma unroll
      for (int c = 0; c < 3; ++c) {
        const float num  = s[bb * 9 + c * 3 + 0];
        const float den1 = s[bb * 9 + c * 3 + 1];
        const float den2 = s[bb * 9 + c * 3 + 2];
        loss += ((1.0f + BETA) * num + EPS) / (den1 + BETA * den2 + EPS);
      }
    }
    // -mean_b( sum_c score / 3 )  with B=2  =>  -(sum of all 6 scores)/6
    out[0] = -loss * (1.0f / 6.0f);
  }
}

extern "C" void kernel_launch(void* const* d_in, const int* in_sizes, int n_in,
                              void* d_out, int out_size, void* d_ws, size_t ws_size,
                              hipStream_t stream) {
  const float* output = (const float*)d_in[0];  // (2,4,64,256,256) f32
  const int*   labels = (const int*)d_in[1];    // (2,2,64,256,256) i32
  float* out     = (float*)d_out;               // scalar f32
  float* partial = (float*)d_ws;

  int P = 1024;                                  // partials per batch
  const size_t need = (size_t)2 * 1024 * 9 * sizeof(float);
  if (ws_size < need) {
    P = (int)(ws_size / (2 * 9 * sizeof(float)));
    if (P < 1) P = 1;
  }

  dim3 grid((unsigned)P, 2);
  ske_partial_kernel<<<grid, 256, 0, stream>>>(output, labels, partial, P);
  ske_final_kernel<<<1, 576, 0, stream>>>(partial, P, out);

  (void)in_sizes; (void)n_in; (void)out_size;
}